// RadialSpectrumFeatures_41583873360196
// MI455X (gfx1250) — compile-verified
//
#include <hip/hip_runtime.h>
#include <cstdint>

// ---------------------------------------------------------------------------
// RadialSpectrumFeatures for MI455X (gfx1250, wave32)
//
// out[pos[c], off_l + n*4 + s] += r^l * fc(r) * exp(-(n+1) r^2/25)
// pos[c] = stable-counting-sort rank of center c by its species, computed
// with V_WMMA_F32_16X16X4_F32 on the one-hot species matrix (rank via
// triangular row-sums of S·S^T, accumulated in the WMMA C operand).
//
// Scatter phase is L2-atomic bound (72M f32 atomics into a 57.6MB
// accumulator resident in the 192MB L2); inputs staged per-tile via the
// CDNA5 async global->LDS path (ASYNCcnt).
// ---------------------------------------------------------------------------

#define N_SPECIES   4
#define FEAT        144
#define CS_CHUNK    1024
#define CS_THREADS  256
#define TILE        256

typedef __attribute__((ext_vector_type(2))) float v2f;
typedef __attribute__((ext_vector_type(8))) float v8f;

// ---- CDNA5 async global->LDS helpers (inline asm; toolchain-portable) -----
__device__ __forceinline__ void async_copy_b128(unsigned lds_byte_addr, const void* gaddr) {
#if defined(__HIP_DEVICE_COMPILE__)
    asm volatile("global_load_async_to_lds_b128 %0, %1, off"
                 :: "v"(lds_byte_addr), "v"(gaddr) : "memory");
#endif
}
__device__ __forceinline__ void wait_asynccnt0() {
#if defined(__HIP_DEVICE_COMPILE__)
    asm volatile("s_wait_asynccnt 0" ::: "memory");
#endif
}
// Low 32 bits of a generic LDS pointer are the LDS byte offset (aperture is
// in addr[63:32]).
__device__ __forceinline__ unsigned lds_off(const void* p) {
    return (unsigned)(uintptr_t)p;
}

// ---- 1) per-chunk species histogram ---------------------------------------
__global__ __launch_bounds__(CS_THREADS)
void rsf_hist_kernel(const int* __restrict__ species, int n, int* __restrict__ blockCounts) {
    __shared__ int cnt[N_SPECIES];
    int t = threadIdx.x;
    if (t < N_SPECIES) cnt[t] = 0;
    __syncthreads();
    int base = blockIdx.x * CS_CHUNK;
    for (int i = t; i < CS_CHUNK; i += blockDim.x) {
        int g = base + i;
        if (g < n) atomicAdd(&cnt[species[g] & 3], 1);
    }
    __syncthreads();
    if (t < N_SPECIES) blockCounts[blockIdx.x * N_SPECIES + t] = cnt[t];
}

// ---- 2) tiny scan: absolute start offset of each (block, species) ---------
__global__ void rsf_scan_kernel(const int* __restrict__ blockCounts,
                                int* __restrict__ blockOffsets, int numBlocks) {
    __shared__ int totals[N_SPECIES];
    int s = threadIdx.x;
    if (s < N_SPECIES) {
        int run = 0;
        for (int b = 0; b < numBlocks; ++b) {
            blockOffsets[b * N_SPECIES + s] = run;
            run += blockCounts[b * N_SPECIES + s];
        }
        totals[s] = run;
    }
    __syncthreads();
    if (s == 0) {
        int base = 0;
        for (int sp = 0; sp < N_SPECIES; ++sp) { int t = totals[sp]; totals[sp] = base; base += t; }
    }
    __syncthreads();
    if (s < N_SPECIES) {
        int b0 = totals[s];
        for (int b = 0; b < numBlocks; ++b) blockOffsets[b * N_SPECIES + s] += b0;
    }
}

// ---- 3) stable rank via WMMA on one-hot species matrix --------------------
// Chunk of 1024 elements = 64 row-tiles of 16. For I-tile, accumulate over
// all J<I:  C += A_I[16x4] x B_J[4x16]  (V_WMMA_F32_16X16X4_F32), where
// A/B are one-hot(species). Row-sum of C = #earlier-full-tile elements with
// same species; in-tile strict-lower handled by a <=15-step LDS scan.
// Fragment layouts (ISA 7.12.2, f32): VGPR v, lane-half h -> K = v + 2h;
// A: M = lane%16; B: N = lane%16. C/D: VGPR v, half h -> M = v + 8h, N = lane%16.
__global__ __launch_bounds__(CS_THREADS)
void rsf_pos_wmma_kernel(const int* __restrict__ species, int n,
                         const int* __restrict__ blockOffsets, int* __restrict__ pos) {
    __shared__ int sp[CS_CHUNK];
    const int t = threadIdx.x;
    const int base = blockIdx.x * CS_CHUNK;
    for (int i = t; i < CS_CHUNK; i += CS_THREADS) {
        int g = base + i;
        sp[i] = (g < n) ? (species[g] & 3) : -1;   // -1 pads -> all-zero one-hot
    }
    __syncthreads();

    const int wave = t >> 5;           // 0..7
    const int lane = t & 31;
    const int m    = lane & 15;        // row (A) / col (B) within tile
    const int kb   = (lane >> 4) << 1; // K base: 0 (lanes 0-15) or 2 (lanes 16-31)

    // strided I-tile assignment for load balance (tile I costs I wmmas)
    for (int ii = 0; ii < 8; ++ii) {
        const int I = wave + (ii << 3);            // 0..63

        const int spI = sp[I * 16 + m];
        v2f a;
        a.x = (spI == kb)     ? 1.0f : 0.0f;
        a.y = (spI == kb + 1) ? 1.0f : 0.0f;

        v8f c = {0.f, 0.f, 0.f, 0.f, 0.f, 0.f, 0.f, 0.f};
        for (int J = 0; J < I; ++J) {              // uniform per wave
            const int spJ = sp[J * 16 + m];
            v2f b;
            b.x = (spJ == kb)     ? 1.0f : 0.0f;
            b.y = (spJ == kb + 1) ? 1.0f : 0.0f;
            c = __builtin_amdgcn_wmma_f32_16x16x4_f32(
                    /*neg_a=*/false, a, /*neg_b=*/false, b,
                    /*c_mod=*/(short)0, c, /*reuse_a=*/false, /*reuse_b=*/false);
        }

        // row-sum over N: reduce the 16 lanes of each half (xor masks < 16)
        #pragma unroll
        for (int v = 0; v < 8; ++v) {
            float x = c[v];
            x += __shfl_xor(x, 1, 32);
            x += __shfl_xor(x, 2, 32);
            x += __shfl_xor(x, 4, 32);
            x += __shfl_xor(x, 8, 32);
            c[v] = x;   // lane half h now holds rowsum(M = v + 8h)
        }
        float oth[8];
        #pragma unroll
        for (int v = 0; v < 8; ++v) oth[v] = __shfl_xor(c[v], 16, 32);

        // lanes 0..15 emit element m of tile I (M = m)
        const int g = base + I * 16 + m;
        if (lane < 16 && g < n) {
            const int v = m & 7;
            float own_v = 0.f, oth_v = 0.f;
            #pragma unroll
            for (int q = 0; q < 8; ++q) {
                if (q == v) { own_v = c[q]; oth_v = oth[q]; }
            }
            const float full = (m < 8) ? own_v : oth_v;   // M<8 in own half, M>=8 in other

            const int mys = sp[I * 16 + m];
            int dcnt = 0;
            for (int j = 0; j < m; ++j) dcnt += (sp[I * 16 + j] == mys) ? 1 : 0;

            pos[g] = blockOffsets[blockIdx.x * N_SPECIES + mys] + (int)(full + 0.5f) + dcnt;
        }
    }
}

// ---- 4) zero the output (harness poisons it) ------------------------------
__global__ void rsf_zero_kernel(float4* __restrict__ out4, long long n4,
                                float* __restrict__ out, long long n) {
    long long i = (long long)blockIdx.x * blockDim.x + threadIdx.x;
    if (i < n4) out4[i] = make_float4(0.f, 0.f, 0.f, 0.f);
    if (i == 0) for (long long k = n4 * 4; k < n; ++k) out[k] = 0.f;
}

// ---- 5) main scatter: async-stage inputs, compute 36 basis values, atomics -
__global__ __launch_bounds__(TILE)
void rsf_scatter_kernel(const float* __restrict__ dirs,
                        const int*   __restrict__ cent,
                        const int*   __restrict__ spec,
                        const int*   __restrict__ pos,
                        float*       __restrict__ out, int P) {
    __shared__ float s_dir[TILE * 3];   // 3072 B
    __shared__ int   s_cent[TILE];      // 1024 B
    __shared__ int   s_spec[TILE];      // 1024 B

    const int t = threadIdx.x;
    const long long tileBase = (long long)blockIdx.x * TILE;

    // Async global->LDS staging; totals (P*12, P*4) are multiples of 16B.
    {
        const long long totalDirB = (long long)P * 12;
        if (t < (TILE * 12) / 16) {                       // 192 lanes x 16B
            long long gb = tileBase * 12 + (long long)t * 16;
            if (gb < totalDirB)
                async_copy_b128(lds_off(&s_dir[t * 4]), (const char*)dirs + gb);
        }
        const long long totalIdxB = (long long)P * 4;
        if (t < (TILE * 4) / 16) {                        // 64 lanes x 16B each stream
            long long gb = tileBase * 4 + (long long)t * 16;
            if (gb < totalIdxB) {
                async_copy_b128(lds_off(&s_cent[t * 4]), (const char*)cent + gb);
                async_copy_b128(lds_off(&s_spec[t * 4]), (const char*)spec + gb);
            }
        }
    }
    wait_asynccnt0();
    __syncthreads();

    long long p = tileBase + t;
    if (p >= P) return;

    float x = s_dir[t * 3 + 0];
    float y = s_dir[t * 3 + 1];
    float z = s_dir[t * 3 + 2];
    float r2 = fmaf(x, x, fmaf(y, y, z * z));
    float r  = sqrtf(r2);
    if (r >= 5.0f) return;                       // fc == 0 -> contributes zeros

    float fc = 0.5f * (__cosf(r * 0.62831853071795864769f) + 1.0f); // cos(pi*r/5)
    float e  = __expf(r2 * -0.04f);                                 // exp(-r^2/25)

    int row = pos[s_cent[t]];
    float* q = out + (size_t)row * FEAT + (s_spec[t] & 3);

    constexpr int nmax[4] = {12, 10, 8, 6};
    constexpr int offs[4] = {0, 48, 88, 120};
    float rl = 1.0f;                              // r^l (l=0 -> 1, matches 0**0)
    #pragma unroll
    for (int l = 0; l < 4; ++l) {
        float v = fc * rl * e;                    // fc * r^l * e^(n+1), n=0
        float* ql = q + offs[l];
        #pragma unroll
        for (int n = 0; n < 12; ++n) {
            if (n < nmax[l]) {
                unsafeAtomicAdd(ql + 4 * n, v);   // global_atomic_add_f32
                v *= e;
            }
        }
        rl *= r;
    }
}

// ---------------------------------------------------------------------------
extern "C" void kernel_launch(void* const* d_in, const int* in_sizes, int n_in,
                              void* d_out, int out_size, void* d_ws, size_t ws_size,
                              hipStream_t stream) {
    const float* dirs = (const float*)d_in[0];   // [P,3] f32
    const int*   pci  = (const int*)d_in[1];     // [P]
    const int*   nsi  = (const int*)d_in[2];     // [P]
    const int*   csi  = (const int*)d_in[3];     // [N_CENTERS]

    const int P  = in_sizes[0] / 3;
    const int nc = in_sizes[3];
    float* out = (float*)d_out;

    const int numBlocks = (nc + CS_CHUNK - 1) / CS_CHUNK;

    // workspace layout (ints): blockCounts | blockOffsets | pos
    int* blockCounts  = (int*)d_ws;
    int* blockOffsets = blockCounts + (size_t)numBlocks * N_SPECIES;
    int* pos          = blockOffsets + (size_t)numBlocks * N_SPECIES;

    rsf_hist_kernel<<<numBlocks, CS_THREADS, 0, stream>>>(csi, nc, blockCounts);
    rsf_scan_kernel<<<1, 64, 0, stream>>>(blockCounts, blockOffsets, numBlocks);
    rsf_pos_wmma_kernel<<<numBlocks, CS_THREADS, 0, stream>>>(csi, nc, blockOffsets, pos);

    long long n  = out_size;
    long long n4 = n / 4;
    int zblocks = (int)((n4 + 255) / 256);
    if (zblocks < 1) zblocks = 1;
    rsf_zero_kernel<<<zblocks, 256, 0, stream>>>((float4*)d_out, n4, out, n);

    int sblocks = (P + TILE - 1) / TILE;
    rsf_scatter_kernel<<<sblocks, TILE, 0, stream>>>(dirs, pci, nsi, pos, out, P);
}